// FastGaussianModel_87376814670608
// MI455X (gfx1250) — compile-verified
//
#include <hip/hip_runtime.h>

typedef float v2f __attribute__((ext_vector_type(2)));
typedef float v8f __attribute__((ext_vector_type(8)));

#define MPTS   50000
#define NG     1024
#define NT_M   (MPTS / 16)   // 3125 M-tiles (exact)
#define NT_N   (NG / 16)     // 64 N-tiles
#define BPAD   1040          // padded pair-row length (bank-conflict avoidance)
#define EPS_F  1e-6f
#define WAVES_PER_BLOCK 8

// -0.5 * log2(e): folded into the B features so the WMMA output is directly
// the argument of exp2 (saves one v_pk_mul_f32 per result pair in the hot loop).
#define NEG_HALF_LOG2E (-0.7213475204444817f)

// ---------------------------------------------------------------------------
// Prep: build the 8xN feature matrix B for  exp2arg[m,n] = A[m,:] . B[:,n]
//   A rows (per point m): [p0^2, p1^2, p2^2, p0, p1, p2, 1, 0]
//   B cols (per gauss n): C * [w0, w1, w2, -2q0w0, -2q1w1, -2q2w2, sum(q^2 w), 0]
//   with C = -0.5*log2(e), so gauss = exp2(A.B) with no post-scale.
// Stored pair-interleaved as float2 so the main kernel uses ds_load_b64:
//   pair j, entry n holds rows (2j, 2j+1).
// ---------------------------------------------------------------------------
__global__ void gm_prep_kernel(const float* __restrict__ positions,
                               const float* __restrict__ log_scales,
                               float* __restrict__ ws) {
    int n = blockIdx.x * blockDim.x + threadIdx.x;
    if (n >= NG) return;

    float q0 = positions[n * 3 + 0];
    float q1 = positions[n * 3 + 1];
    float q2 = positions[n * 3 + 2];
    float e0 = __expf(log_scales[n * 3 + 0]);
    float e1 = __expf(log_scales[n * 3 + 1]);
    float e2 = __expf(log_scales[n * 3 + 2]);
    float w0 = 1.0f / (e0 * e0 + EPS_F);
    float w1 = 1.0f / (e1 * e1 + EPS_F);
    float w2 = 1.0f / (e2 * e2 + EPS_F);
    float sq = q0 * q0 * w0 + q1 * q1 * w1 + q2 * q2 * w2;

    const float C = NEG_HALF_LOG2E;
    float2* pB = (float2*)ws;
    pB[0 * BPAD + n] = make_float2(C * w0, C * w1);
    pB[1 * BPAD + n] = make_float2(C * w2, C * -2.0f * q0 * w0);
    pB[2 * BPAD + n] = make_float2(C * -2.0f * q1 * w1, C * -2.0f * q2 * w2);
    pB[3 * BPAD + n] = make_float2(C * sq, 0.0f);
}

// ---------------------------------------------------------------------------
// Main: one wave32 per 16-row M-tile. Per N-tile: 2x v_wmma_f32_16x16x4_f32
// (K=8 rank factorization of the quadratic form, pre-scaled into exp2 domain),
// then exp2 + weighted accumulate per lane, final 16-lane shfl_xor reduction.
// ---------------------------------------------------------------------------
__launch_bounds__(256)
__global__ void gm_eval_kernel(const float* __restrict__ points,
                               const float* __restrict__ inten,
                               const float* __restrict__ wsB,
                               float* __restrict__ out) {
    __shared__ float2 sB[4 * BPAD];   // 33280 B
    __shared__ float  sI[NG];         //  4096 B

    const int tid = threadIdx.x;

    // Stage B-features + intensities into LDS once per block.
    const float2* gB = (const float2*)wsB;
    for (int i = tid; i < 4 * BPAD; i += 256) sB[i] = gB[i];
    for (int i = tid; i < NG; i += 256)       sI[i] = inten[i];
    __syncthreads();

    const int lane = tid & 31;
    const int h    = lane >> 4;   // half-wave: selects K-pair / C-row half
    const int c    = lane & 15;   // column within tile / row within half

    const int wave  = __builtin_amdgcn_readfirstlane(tid >> 5);
    const int mtile = blockIdx.x * WAVES_PER_BLOCK + wave;
    if (mtile >= NT_M) return;

    // A-operand (16x4 f32 layout: VGPR0 = K(2h), VGPR1 = K(2h+1), M = c)
    const int row = mtile * 16 + c;
    const float px = points[row * 3 + 0];
    const float py = points[row * 3 + 1];
    const float pz = points[row * 3 + 2];

    v2f a1, a2;                       // K0..3 and K4..7 slices
    a1.x = h ? pz * pz : px * px;     // K2 : K0
    a1.y = h ? px      : py * py;     // K3 : K1
    a2.x = h ? 1.0f    : py;          // K6 : K4
    a2.y = h ? 0.0f    : pz;          // K7 : K5

    v8f acc = {};

    for (int nt = 0; nt < NT_N; ++nt) {
        const int n = nt * 16 + c;
        const float2 b1 = sB[h * BPAD + n];        // rows (2h, 2h+1)
        const float2 b2 = sB[(2 + h) * BPAD + n];  // rows (4+2h, 5+2h)
        const float  In = sI[n];

        v2f bb1; bb1.x = b1.x; bb1.y = b1.y;
        v2f bb2; bb2.x = b2.x; bb2.y = b2.y;

        v8f cz = {};
        v8f t = __builtin_amdgcn_wmma_f32_16x16x4_f32(
            false, a1, false, bb1, (short)0, cz, false, false);
        v8f q = __builtin_amdgcn_wmma_f32_16x16x4_f32(
            false, a2, false, bb2, (short)0, t, false, false);

        #pragma unroll
        for (int j = 0; j < 8; ++j) {
            float g = __builtin_amdgcn_exp2f(q[j]);   // pre-scaled by WMMA
            acc[j] += g * In;
        }
    }

    // C/D layout: VGPR j holds row (j + 8h), column = c. Sum over the 16
    // lanes of each half-wave (columns) with xor-shuffles confined to width 16.
    #pragma unroll
    for (int j = 0; j < 8; ++j) {
        float v = acc[j];
        v += __shfl_xor(v, 1, 16);
        v += __shfl_xor(v, 2, 16);
        v += __shfl_xor(v, 4, 16);
        v += __shfl_xor(v, 8, 16);
        acc[j] = v;
    }

    if (c == 0) {
        const int base = mtile * 16 + h * 8;
        #pragma unroll
        for (int j = 0; j < 8; ++j) out[base + j] = acc[j];
    }
}

// ---------------------------------------------------------------------------
extern "C" void kernel_launch(void* const* d_in, const int* in_sizes, int n_in,
                              void* d_out, int out_size, void* d_ws, size_t ws_size,
                              hipStream_t stream) {
    const float* points     = (const float*)d_in[0];
    const float* positions  = (const float*)d_in[1];
    const float* log_scales = (const float*)d_in[2];
    const float* inten      = (const float*)d_in[3];
    float*       out        = (float*)d_out;
    float*       ws         = (float*)d_ws;   // needs 4*BPAD*2*4 = 33280 bytes

    gm_prep_kernel<<<dim3((NG + 255) / 256), dim3(256), 0, stream>>>(
        positions, log_scales, ws);

    const int blocks = (NT_M + WAVES_PER_BLOCK - 1) / WAVES_PER_BLOCK;  // 391
    gm_eval_kernel<<<dim3(blocks), dim3(256), 0, stream>>>(
        points, inten, ws, out);
}